// Generator_78924319031843
// MI455X (gfx1250) — compile-verified
//
#include <hip/hip_runtime.h>
#include <hip/hip_bf16.h>
#include <math.h>

// CDNA5 fp32 WMMA fragments: A(16x4)=2 VGPR, B(4x16)=2 VGPR, C/D(16x16)=8 VGPR
typedef __attribute__((ext_vector_type(2))) float v2f;
typedef __attribute__((ext_vector_type(8))) float v8f;

#define ROWS       16      // batch rows per workgroup
#define LDST       132     // LDS row stride (floats), padded to avoid bank conflicts
#define TZK        68      // [t, z] width 65 padded up to multiple of 4
#define HID        64
#define TSTEPS     64
#define NSTEP      63
#define BATCH_N    4096
#define NOISE_DIM  32
#define BLOCK      256
#define NWAVE      8

struct SdeParams {
  const float* ts;
  const float* noise;
  const float* dWg;                      // (63, 4096, 32)
  const float* iW[4]; const float* ib[4];  // initial MLP
  const float* fW[4]; const float* fb[4];  // drift MLP
  const float* gW[4]; const float* gb[4];  // diffusion MLP
  const float* rW; const float* rb;        // readout
  float* out;                              // (4096, 64, 16)
};

__device__ __forceinline__ float fast_rcp(float x) {
#if __has_builtin(__builtin_amdgcn_rcpf)
  return __builtin_amdgcn_rcpf(x);         // v_rcp_f32 (TRANS pipe)
#else
  return 1.0f / x;
#endif
}

__device__ __forceinline__ float fast_tanh(float x) {
#if __has_builtin(__builtin_amdgcn_tanhf)
  return __builtin_amdgcn_tanhf(x);        // v_tanh_f32 (gfx1250 TRANS op)
#elif __has_builtin(__builtin_amdgcn_tanh_f32)
  return __builtin_amdgcn_tanh_f32(x);
#else
  return tanhf(x);
#endif
}

__device__ __forceinline__ float lipswish(float x) {
  // 0.909 * silu(x) = 0.909 * x * sigmoid(x)
  return 0.909f * x * fast_rcp(1.0f + __expf(-x));
}

// One MLP layer on the workgroup's 16-row tile using V_WMMA_F32_16X16X4_F32.
// OUT[r][o] = act( sum_k IN[r][k] * W[o][k] + bias[o] )
// act: 0=identity, 1=lipswish, 2=tanh
__device__ __forceinline__ void gemm_layer(
    const float* __restrict__ W, const float* __restrict__ bias,
    const float* __restrict__ IN, float* __restrict__ OUT,
    int Odim, int Kdim, int Kpad, int outStride, int act)
{
  const int tid  = threadIdx.x;
  const int wave = tid >> 5;
  const int lane = tid & 31;
  const int half = lane >> 4;   // 0: lanes 0-15, 1: lanes 16-31
  const int l15  = lane & 15;
  const int ntiles = Odim >> 4; // 16x16 output tiles (single row-tile: ROWS==16)

  for (int tile = wave; tile < ntiles; tile += NWAVE) {
    const int o0 = tile << 4;
    const int oc = o0 + l15;
    const float bv = bias[oc];
    v8f acc = {bv, bv, bv, bv, bv, bv, bv, bv};
    const float* arow = IN + l15 * LDST;             // A row M = l15
    const float* wrow = W + (size_t)oc * Kdim;       // B col N = l15 -> weight row
    for (int k = 0; k < Kpad; k += 4) {
      const int ka = k + 2 * half;                   // lanes 0-15 hold K={k,k+1}, 16-31 hold K={k+2,k+3}
      v2f a; a.x = arow[ka]; a.y = arow[ka + 1];     // IN is zero-padded to Kpad
      const int i0 = (ka      < Kdim) ? ka      : 0; // clamp pad columns (A is zero there)
      const int i1 = (ka + 1  < Kdim) ? ka + 1  : 0;
      v2f b; b.x = wrow[i0]; b.y = wrow[i1];
      acc = __builtin_amdgcn_wmma_f32_16x16x4_f32(false, a, false, b,
                                                  (short)0, acc, false, false);
    }
    // D layout: VGPR r -> M = r + 8*half, N = l15
    #pragma unroll
    for (int r = 0; r < 8; ++r) {
      float v = acc[r];
      if (act == 1)      v = lipswish(v);
      else if (act == 2) v = fast_tanh(v);
      OUT[(r + 8 * half) * outStride + o0 + l15] = v;
    }
  }
}

// f, g = MLPs(concat(t, S)); writes f to fOut, and contracts g with dwA (and dwB)
// without ever materializing the full (ROWS,64,32) g tensor.
__device__ void f_and_g(const SdeParams& P, float t,
                        const float* __restrict__ S,
                        float* TZ, float* A, float* B,
                        float* DWA, float* DWB,
                        float* fOut, float* gA, float* gB,
                        const float* dwAg, const float* dwBg, int hasB)
{
  const int tid = threadIdx.x;
  // Build [t, z, 0-pad] input tile (K = 65 padded to 68)
  for (int i = tid; i < ROWS * TZK; i += BLOCK) {
    const int r = i / TZK, c = i - r * TZK;
    float v = 0.0f;
    if (c == 0)        v = t;
    else if (c <= HID) v = S[r * HID + (c - 1)];
    TZ[r * LDST + c] = v;
  }
  // Stage this tile's dW rows in LDS
  for (int i = tid; i < ROWS * NOISE_DIM; i += BLOCK) {
    const int r = i >> 5, n = i & 31;
    DWA[i] = dwAg[(size_t)r * NOISE_DIM + n];
    if (hasB) DWB[i] = dwBg[(size_t)r * NOISE_DIM + n];
  }
  __syncthreads();

  // drift MLP -> fOut (tanh)
  gemm_layer(P.fW[0], P.fb[0], TZ, A, 128, 65, TZK, LDST, 1); __syncthreads();
  gemm_layer(P.fW[1], P.fb[1], A,  B, 128, 128, 128, LDST, 1); __syncthreads();
  gemm_layer(P.fW[2], P.fb[2], B,  A, 128, 128, 128, LDST, 1); __syncthreads();
  gemm_layer(P.fW[3], P.fb[3], A,  fOut, 64, 128, 128, HID, 2); __syncthreads();

  // diffusion MLP trunk
  gemm_layer(P.gW[0], P.gb[0], TZ, A, 128, 65, TZK, LDST, 1); __syncthreads();
  gemm_layer(P.gW[1], P.gb[1], A,  B, 128, 128, 128, LDST, 1); __syncthreads();
  gemm_layer(P.gW[2], P.gb[2], B,  A, 128, 128, 128, LDST, 1); __syncthreads();

  // final 128->2048 layer in 16 chunks of 128 cols; fuse tanh + (g.dw) contraction
  for (int chunk = 0; chunk < 16; ++chunk) {
    gemm_layer(P.gW[3] + (size_t)chunk * 128 * 128, P.gb[3] + chunk * 128,
               A, B, 128, 128, 128, LDST, 2);
    __syncthreads();

    // Prefetch next chunk's 64KB weight block into cache (global_prefetch_b8)
    if (chunk + 1 < 16) {
      const char* nw = (const char*)(P.gW[3] + (size_t)(chunk + 1) * 128 * 128);
      __builtin_prefetch(nw + tid * 256, 0, 0);
    }

    // chunk covers h = chunk*4 .. chunk*4+3 with full n = 0..31 each
    for (int i = tid; i < ROWS * 4; i += BLOCK) {
      const int b = i >> 2, hl = i & 3;
      const float* grow = B + b * LDST + hl * 32;
      const float* da = DWA + b * 32;
      const float* db = DWB + b * 32;
      float sA = 0.0f, sB = 0.0f;
      #pragma unroll 8
      for (int n = 0; n < 32; ++n) {
        const float gv = grow[n];
        sA += gv * da[n];
        if (hasB) sB += gv * db[n];
      }
      gA[b * HID + chunk * 4 + hl] = sA;
      if (hasB) gB[b * HID + chunk * 4 + hl] = sB;
    }
    __syncthreads();
  }
}

__device__ void readout_t(const SdeParams& P, const float* __restrict__ y,
                          int b0, int t) {
  const int tid = threadIdx.x;
  for (int i = tid; i < ROWS * 16; i += BLOCK) {
    const int b = i >> 4, d = i & 15;
    const float* wr = P.rW + d * HID;
    const float* yr = y + b * HID;
    float acc = 0.0f;
    #pragma unroll 8
    for (int h = 0; h < HID; ++h) acc += yr[h] * wr[h];
    P.out[((size_t)(b0 + b) * TSTEPS + t) * 16 + d] = P.rb[d] + acc;
  }
}

__global__ __launch_bounds__(BLOCK) void sde_reversible_heun_kernel(SdeParams P) {
  __shared__ float TZ[ROWS * LDST];
  __shared__ float A [ROWS * LDST];
  __shared__ float B [ROWS * LDST];
  __shared__ float st[7][ROWS * HID];      // y, z, f, f1, g rotation x3
  __shared__ float DWA[ROWS * NOISE_DIM];
  __shared__ float DWB[ROWS * NOISE_DIM];

  const int tid = threadIdx.x;
  const int b0  = blockIdx.x * ROWS;

  float* y  = st[0];
  float* z  = st[1];
  float* fP = st[2];   // f at current step
  float* fN = st[3];   // f1
  float* gP = st[4];   // g_k . dw_k   (this step's gdw)
  float* gC = st[5];   // g_{k+1} . dw_k
  float* gN = st[6];   // g_{k+1} . dw_{k+1} (next step's gdw)

  // ---- initial MLP: noise(32) -> 128 -> 128 -> 128 -> 64 ----
  for (int i = tid; i < ROWS * NOISE_DIM; i += BLOCK) {
    const int r = i >> 5, c = i & 31;
    A[r * LDST + c] = P.noise[(size_t)(b0 + r) * NOISE_DIM + c];
  }
  __syncthreads();
  gemm_layer(P.iW[0], P.ib[0], A, B, 128, 32, 32, LDST, 1); __syncthreads();
  gemm_layer(P.iW[1], P.ib[1], B, A, 128, 128, 128, LDST, 1); __syncthreads();
  gemm_layer(P.iW[2], P.ib[2], A, B, 128, 128, 128, LDST, 1); __syncthreads();
  gemm_layer(P.iW[3], P.ib[3], B, y, 64, 128, 128, HID, 0); __syncthreads();

  for (int i = tid; i < ROWS * HID; i += BLOCK) z[i] = y[i];
  readout_t(P, y, b0, 0);
  __syncthreads();

  // initial f_and_g(t0, y0): f0 -> fP; g0 . dW[0] -> gP
  {
    const float t0 = P.ts[0];
    const float* dw0 = P.dWg + (size_t)b0 * NOISE_DIM;
    f_and_g(P, t0, y, TZ, A, B, DWA, DWB, fP, gP, gN, dw0, dw0, 0);
  }
  __syncthreads();

  // ---- reversible Heun scan ----
  for (int k = 0; k < NSTEP; ++k) {
    const float t1 = P.ts[k + 1];
    const float dt = t1 - P.ts[k];
    // z1 = 2y - z + f*dt + gdw
    for (int i = tid; i < ROWS * HID; i += BLOCK)
      z[i] = 2.0f * y[i] - z[i] + fP[i] * dt + gP[i];
    __syncthreads();

    const float* dwA = P.dWg + ((size_t)k * BATCH_N + b0) * NOISE_DIM;
    const int hasB = (k + 1 < NSTEP);
    const float* dwB = P.dWg + ((size_t)(hasB ? k + 1 : k) * BATCH_N + b0) * NOISE_DIM;
    // f1 -> fN; g1.dw_k -> gC; g1.dw_{k+1} -> gN
    f_and_g(P, t1, z, TZ, A, B, DWA, DWB, fN, gC, gN, dwA, dwB, hasB);
    __syncthreads();

    // y1 = y + 0.5*(f+f1)*dt + 0.5*(gdw + g1.dw_k)
    for (int i = tid; i < ROWS * HID; i += BLOCK)
      y[i] += 0.5f * (fP[i] + fN[i]) * dt + 0.5f * (gP[i] + gC[i]);
    __syncthreads();
    readout_t(P, y, b0, k + 1);

    // rotate state for next step
    float* tf = fP; fP = fN; fN = tf;
    float* tg = gP; gP = gN; gN = tg;
    __syncthreads();
  }
}

extern "C" void kernel_launch(void* const* d_in, const int* in_sizes, int n_in,
                              void* d_out, int out_size, void* d_ws, size_t ws_size,
                              hipStream_t stream) {
  // setup_inputs order:
  // 0 ts(64) 1 batch(1,int) 2 initial_noise(4096x32) 3 dW(63x4096x32)
  // 4..11  initial_params  W0,b0,W1,b1,W2,b2,W3,b3
  // 12..19 drift_params    W0,b0,W1,b1,W2,b2,W3,b3
  // 20..27 diffusion_params W0,b0,W1,b1,W2,b2,W3,b3
  // 28 readout_W(16x64) 29 readout_b(16)
  SdeParams P;
  P.ts    = (const float*)d_in[0];
  P.noise = (const float*)d_in[2];
  P.dWg   = (const float*)d_in[3];
  for (int l = 0; l < 4; ++l) {
    P.iW[l] = (const float*)d_in[4  + 2 * l]; P.ib[l] = (const float*)d_in[5  + 2 * l];
    P.fW[l] = (const float*)d_in[12 + 2 * l]; P.fb[l] = (const float*)d_in[13 + 2 * l];
    P.gW[l] = (const float*)d_in[20 + 2 * l]; P.gb[l] = (const float*)d_in[21 + 2 * l];
  }
  P.rW  = (const float*)d_in[28];
  P.rb  = (const float*)d_in[29];
  P.out = (float*)d_out;

  dim3 grid(BATCH_N / ROWS);   // 256 workgroups x 8 waves, fully independent
  sde_reversible_heun_kernel<<<grid, BLOCK, 0, stream>>>(P);
}